// AttentionLayer_87333864996987
// MI455X (gfx1250) — compile-verified
//
#include <hip/hip_runtime.h>

#define KSZ 7
#define PAD 3
#define TW 32
#define TH 32
#define HALO 6
#define TILE_W (TW + HALO)      // 38
#define TILE_H (TH + HALO)      // 38
#define LDS_STRIDE 40           // padded row stride (floats)
#define HW 128
#define BC (4 * 64)

#define HAVE_ASYNC_LDS (__has_builtin(__builtin_amdgcn_global_load_async_to_lds_b32) && \
                        __has_builtin(__builtin_amdgcn_s_wait_asynccnt))

typedef __attribute__((address_space(1))) int* gptr_i32;
typedef __attribute__((address_space(3))) int* lptr_i32;

#define LOG2E 1.44269504088896340736f

__global__ __launch_bounds__(256) void natt7x7_softmax_kernel(
    const float* __restrict__ x, float* __restrict__ out) {
  __shared__ float tile[TILE_H * LDS_STRIDE];

  const int tx = threadIdx.x & 31;        // column within tile (0..31)
  const int ty = threadIdx.x >> 5;        // wave id (0..7)
  const int tile_w0 = blockIdx.x * TW;
  const int tile_h0 = blockIdx.y * TH;
  const int bc = blockIdx.z;              // fused (b,c) plane index

  const float* __restrict__ plane = x + (size_t)bc * (HW * HW);
  float* __restrict__ oplane = out + (size_t)bc * (HW * HW);

  // ---- cooperative halo-tile load: async global->LDS, zeros outside image ----
  const int nelem = TILE_H * TILE_W;      // 1444
  for (int idx = threadIdx.x; idx < nelem; idx += 256) {
    const int r = idx / TILE_W;
    const int c = idx - r * TILE_W;
    const int gr = tile_h0 - PAD + r;
    const int gc = tile_w0 - PAD + c;
    const bool inb = (gr >= 0) & (gr < HW) & (gc >= 0) & (gc < HW);
#if HAVE_ASYNC_LDS
    if (inb) {
      __builtin_amdgcn_global_load_async_to_lds_b32(
          (gptr_i32)(plane + gr * HW + gc),
          (lptr_i32)&tile[r * LDS_STRIDE + c],
          /*offset=*/0, /*cpol=*/0);
    } else {
      tile[r * LDS_STRIDE + c] = 0.0f;    // disjoint addresses from async writes
    }
#else
    tile[r * LDS_STRIDE + c] = inb ? plane[gr * HW + gc] : 0.0f;
#endif
  }
#if HAVE_ASYNC_LDS
  __builtin_amdgcn_s_wait_asynccnt(0);
#endif
  __syncthreads();

  // ---- per-pixel 7x7 softmax-weighted aggregation ----
#pragma unroll
  for (int pp = 0; pp < 4; ++pp) {
    const int pr = ty + pp * 8;                         // pixel row in tile
    const float* base = &tile[pr * LDS_STRIDE + tx];    // top-left of window

    float v[KSZ * KSZ];
    const float xc = base[3 * LDS_STRIDE + 3];          // center value
    float m = -3.402823466e38f;

#pragma unroll
    for (int i = 0; i < KSZ; ++i) {
#pragma unroll
      for (int j = 0; j < KSZ; ++j) {
        const float n = base[i * LDS_STRIDE + j];
        v[i * KSZ + j] = n;
        m = fmaxf(m, xc * n);              // running max of scores
      }
    }

    float den = 0.0f;
    float num = 0.0f;
#if __has_builtin(__builtin_amdgcn_exp2f)
    // exp(s - m) == exp2(s*log2e - m*log2e): fold log2e in once per pixel,
    // one v_fma + one v_exp per tap.
    const float xl = xc * LOG2E;
    const float ml = m * LOG2E;
#pragma unroll
    for (int t = 0; t < KSZ * KSZ; ++t) {
      const float e = __builtin_amdgcn_exp2f(fmaf(xl, v[t], -ml));
      den += e;
      num = fmaf(e, v[t], num);
    }
#else
#pragma unroll
    for (int t = 0; t < KSZ * KSZ; ++t) {
      const float e = __expf(fmaf(xc, v[t], -m));
      den += e;
      num = fmaf(e, v[t], num);
    }
#endif

#if __has_builtin(__builtin_amdgcn_rcpf)
    const float r = num * __builtin_amdgcn_rcpf(den);   // v_rcp_f32 + v_mul
#else
    const float r = num / den;
#endif
    oplane[(tile_h0 + pr) * HW + (tile_w0 + tx)] = r;
  }
}

extern "C" void kernel_launch(void* const* d_in, const int* in_sizes, int n_in,
                              void* d_out, int out_size, void* d_ws, size_t ws_size,
                              hipStream_t stream) {
  (void)in_sizes; (void)n_in; (void)d_ws; (void)ws_size; (void)out_size;
  const float* x = (const float*)d_in[0];
  float* out = (float*)d_out;
  dim3 grid(HW / TW, HW / TH, BC);       // 4 x 4 x 256 = 4096 blocks
  natt7x7_softmax_kernel<<<grid, 256, 0, stream>>>(x, out);
}